// MLPDecoder_85341000172344
// MI455X (gfx1250) — compile-verified
//
#include <hip/hip_runtime.h>

#define DIMZ 128
#define HIDDEN 128
#define TM 128                 // edges per block
#define SA_STRIDE 132          // padded f32 stride for feature tile (conflict-free frag loads)
#define SB_STRIDE 144          // padded f32 stride for W1 tile (matches TDM pad 16 dw / 128 dw)

typedef float    v2f __attribute__((ext_vector_type(2)));
typedef float    v4f __attribute__((ext_vector_type(4)));
typedef float    v8f __attribute__((ext_vector_type(8)));
typedef unsigned v4u __attribute__((ext_vector_type(4)));
typedef unsigned v8u __attribute__((ext_vector_type(8)));

__global__ __launch_bounds__(256) void mlp_decoder_kernel(
    const float* __restrict__ z,
    const long long* __restrict__ eidx,   // [2, E] int64
    const float* __restrict__ W1,         // [512, 128]
    const float* __restrict__ b1,         // [128]
    const float* __restrict__ W2,         // [128, 1]
    const float* __restrict__ b2,         // [1]
    float* __restrict__ out,              // [E]
    int Etot)
{
    __shared__ float sA[TM * SA_STRIDE];      // feature segment tile [128 rows x 128 k]
    __shared__ float sB[DIMZ * SB_STRIDE];    // W1 segment tile     [128 k x 128 n]
    __shared__ float sOut[TM];
    __shared__ int   sOff[2 * TM];            // src offsets, then dst offsets (float units)

    const int t    = threadIdx.x;
    const int lane = t & 31;
    const int wave = t >> 5;                  // 8 waves: wave owns N-tile [16w, 16w+16)
    const int half = lane >> 4;
    const int l16  = lane & 15;
    const long long base = (long long)blockIdx.x * TM;

    // ---- stage edge indices (clamped; OOB rows masked at store) ----
    {
        int r = t & 127;
        int which = t >> 7;                   // 0 = src row, 1 = dst row
        long long e = base + r;
        if (e >= Etot) e = Etot - 1;
        long long idx = eidx[(long long)which * Etot + e];
        sOff[which * TM + r] = (int)idx * DIMZ;
    }
    if (t < TM) sOut[t] = 0.0f;

    v8f acc[8];
#pragma unroll
    for (int mt = 0; mt < 8; ++mt)
        acc[mt] = v8f{0.f, 0.f, 0.f, 0.f, 0.f, 0.f, 0.f, 0.f};

    const int nCol  = (wave << 4) + l16;      // this lane's N column
    const int khalf = half << 1;              // K sub-offset per ISA A/B lane layout

    const unsigned ldsB = (unsigned)(uintptr_t)(void*)sB;   // LDS byte offset of sB

    // ---- K loop over 4 feature segments: zi | zj | |zi-zj| | zi*zj ----
    for (int s = 0; s < 4; ++s) {
        __syncthreads();   // prior GEMM reads of sA/sB done (also covers sOff/sOut init)

        // ---- wave 0: kick TDM DMA of the W1 segment tile into sB ----
        // D# group0: count=1, lds_addr, global_addr, type=2 ("image")
        // D# group1: data_size=4B, pad_enable, pad_interval=6 (128 dw), pad_amount=15 (16 dw)
        //            tensor_dim0=128, tensor_dim1=512, tile 128x128, dim0_stride=128
        if (wave == 0) {
            unsigned long long ga =
                (unsigned long long)(uintptr_t)(W1 + s * DIMZ * HIDDEN);
            v4u g0;
            g0[0] = 1u;                                   // count=1 valid descriptor
            g0[1] = ldsB;                                 // LDS destination (bytes)
            g0[2] = (unsigned)ga;                         // global_addr[31:0]
            g0[3] = (unsigned)((ga >> 32) & 0x1FFFFFFu)   // global_addr[56:32]
                  | (2u << 30);                           // type=2
            v8u g1;
            g1[0] = (2u << 16)        // data_size = 4 bytes
                  | (1u << 20)        // pad_enable
                  | (6u << 22)        // pad_interval: 128 dwords
                  | (15u << 25);      // pad_amount: 16 dwords -> row stride 144 floats
            g1[1] = (unsigned)(DIMZ) << 16;               // tensor_dim0[15:0] = 128
            g1[2] = (unsigned)(4 * DIMZ) << 16;           // tensor_dim1[15:0] = 512
            g1[3] = (unsigned)(DIMZ) << 16;               // tile_dim0 = 128
            g1[4] = (unsigned)(DIMZ);                     // tile_dim1 = 128, tile_dim2 = 0
            g1[5] = (unsigned)(HIDDEN);                   // tensor_dim0_stride = 128
            g1[6] = 0u;
            g1[7] = 0u;
            asm volatile("tensor_load_to_lds %0, %1" :: "s"(g0), "s"(g1) : "memory");
        }

        // ---- all threads: stage feature segment tile into sA (L2-resident gathers) ----
#pragma unroll
        for (int i = 0; i < 16; ++i) {
            int linear = t + 256 * i;         // 4096 float4 slots = 128 rows x 32 groups
            int row = linear >> 5;
            int c   = (linear & 31) << 2;
            const float* zi = z + sOff[row];
            const float* zj = z + sOff[TM + row];
            v4f v;
            if (s == 0) {
                v = *(const v4f*)(zi + c);
            } else if (s == 1) {
                v = *(const v4f*)(zj + c);
            } else {
                v4f a = *(const v4f*)(zi + c);
                v4f b = *(const v4f*)(zj + c);
                if (s == 2) {
                    v[0] = fabsf(a[0] - b[0]); v[1] = fabsf(a[1] - b[1]);
                    v[2] = fabsf(a[2] - b[2]); v[3] = fabsf(a[3] - b[3]);
                } else {
                    v[0] = a[0] * b[0]; v[1] = a[1] * b[1];
                    v[2] = a[2] * b[2]; v[3] = a[3] * b[3];
                }
            }
            *(v4f*)&sA[row * SA_STRIDE + c] = v;
        }

        if (wave == 0)
            __builtin_amdgcn_s_wait_tensorcnt(0);   // TDM done before the barrier
        __syncthreads();

        // ---- GEMM: 32 k-steps x 8 M-tiles of v_wmma_f32_16x16x4_f32 ----
        for (int k0 = 0; k0 < DIMZ; k0 += 4) {
            const int kk = k0 + khalf;
            v2f bfrag;
            bfrag[0] = sB[kk * SB_STRIDE + nCol];
            bfrag[1] = sB[(kk + 1) * SB_STRIDE + nCol];
#pragma unroll
            for (int mt = 0; mt < 8; ++mt) {
                v2f afrag = *(const v2f*)&sA[(mt * 16 + l16) * SA_STRIDE + kk];
                acc[mt] = __builtin_amdgcn_wmma_f32_16x16x4_f32(
                    false, afrag, false, bfrag, (short)0, acc[mt], false, false);
            }
        }
    }

    // ---- epilogue: bias + ReLU + dot(W2), wave32 butterfly + LDS reduce ----
    __syncthreads();
    const float b1v = b1[nCol];
    const float w2v = W2[nCol];
#pragma unroll
    for (int mt = 0; mt < 8; ++mt) {
#pragma unroll
        for (int j = 0; j < 8; ++j) {
            float h = acc[mt][j] + b1v;
            h = h > 0.f ? h : 0.f;
            float p = h * w2v;
#pragma unroll
            for (int m = 1; m < 16; m <<= 1)
                p += __shfl_xor(p, m, 32);    // reduce 16 N-lanes within each half
            if (l16 == 0) {
                int row = mt * 16 + j + (half << 3);
                atomicAdd(&sOut[row], p);     // ds_add_f32 across waves
            }
        }
    }
    __syncthreads();

    if (t < TM) {
        long long e = base + t;
        if (e < Etot) out[e] = sOut[t] + b2[0];
    }
}

extern "C" void kernel_launch(void* const* d_in, const int* in_sizes, int n_in,
                              void* d_out, int out_size, void* d_ws, size_t ws_size,
                              hipStream_t stream) {
    const float*     z    = (const float*)d_in[0];
    const long long* eidx = (const long long*)d_in[1];
    const float*     W1   = (const float*)d_in[2];
    const float*     b1   = (const float*)d_in[3];
    const float*     W2   = (const float*)d_in[4];
    const float*     b2   = (const float*)d_in[5];
    float*           out  = (float*)d_out;

    const int Etot = in_sizes[1] / 2;         // edge_label_index is [2, E]
    if (Etot <= 0) return;
    const int grid = (Etot + TM - 1) / TM;
    mlp_decoder_kernel<<<grid, 256, 0, stream>>>(z, eidx, W1, b1, W2, b2, out, Etot);
}